// MatchingLoss_80900003987883
// MI455X (gfx1250) — compile-verified
//
#include <hip/hip_runtime.h>
#include <hip/hip_bf16.h>
#include <stdint.h>

// ---------------------------------------------------------------------------
// MatchingLoss on MI455X (gfx1250, wave32)
//   per-batch: sum_i ( |proj_i - img_i| - |org_i - img_i| )   (one accumulator)
//   finalize : diff = sum/(N*1440); reward/penalty; 1000*mean over 64 batches
// Data path : global_load_async_to_lds_b128 (ASYNCcnt), wave-private double
//             buffering in LDS. Wave reduction via V_WMMA_F32_16X16X4_F32.
// ---------------------------------------------------------------------------

#define NBATCH      64
#define NPTS        100000
#define GROUPS      50000        // float4 groups per batch (2 points each)
#define BLK_PER_B   32
#define THREADS     256
#define WAVES       8
#define WPB         (BLK_PER_B * WAVES)   // 256 waves per batch
#define FULL_TILES  1562                  // 32-group tiles fully in range
#define TILES       1563                  // last tile has 16 groups
#define DENOM       1440.0f

typedef float v2f __attribute__((ext_vector_type(2)));
typedef float v8f __attribute__((ext_vector_type(8)));

__device__ static inline void async_cp16(unsigned lds_off, const void* gptr) {
    // GLOBAL_LOAD_ASYNC_TO_LDS_B128, GV mode (64-bit vaddr, saddr=off).
    asm volatile("global_load_async_to_lds_b128 %0, %1, off"
                 :: "v"(lds_off), "v"((unsigned long long)(uintptr_t)gptr)
                 : "memory");
}

__device__ static inline float pair_diff(float4 i, float4 p, float4 o) {
    float dx0 = p.x - i.x, dy0 = p.y - i.y;
    float dx1 = p.z - i.z, dy1 = p.w - i.w;
    float ex0 = o.x - i.x, ey0 = o.y - i.y;
    float ex1 = o.z - i.z, ey1 = o.w - i.w;
    return (__fsqrt_rn(dx0 * dx0 + dy0 * dy0) + __fsqrt_rn(dx1 * dx1 + dy1 * dy1))
         - (__fsqrt_rn(ex0 * ex0 + ey0 * ey0) + __fsqrt_rn(ex1 * ex1 + ey1 * ey1));
}

__global__ __launch_bounds__(THREADS)
void matchloss_partial_kernel(const float4* __restrict__ img,
                              const float4* __restrict__ proj,
                              const float4* __restrict__ org,
                              float* __restrict__ partials) {
    // [wave][buf][array][lane] : 8 * 2 * 3 * 32 * 16B = 24 KB
    __shared__ __align__(16) float4 stage[WAVES][2][3][32];
    __shared__ float wsum[WAVES];

    const unsigned tid   = threadIdx.x;
    const unsigned wave  = tid >> 5;
    const unsigned lane  = tid & 31u;
    const unsigned batch = blockIdx.x >> 5;          // 32 blocks per batch
    const unsigned sblk  = blockIdx.x & 31u;
    const unsigned wq    = sblk * WAVES + wave;      // 0..255 within batch
    const unsigned gbase = batch * (unsigned)GROUPS; // float4 index base

    // Low 32 bits of a generic LDS pointer == LDS byte offset (flat aperture).
    const unsigned ldsBase = (unsigned)(uintptr_t)&stage[wave][0][0][0];

    auto issue_tile = [&](unsigned tt, unsigned buf) {
        const unsigned g     = gbase + tt * 32u + lane;
        const unsigned lbase = ldsBase + buf * (3u * 32u * 16u) + lane * 16u;
        async_cp16(lbase +    0u, img  + g);
        async_cp16(lbase +  512u, proj + g);
        async_cp16(lbase + 1024u, org  + g);
    };

    float acc = 0.0f;
    unsigned t   = wq;
    unsigned cur = 0;
    if (t < FULL_TILES) issue_tile(t, 0);

    while (t < TILES) {
        const unsigned tn = t + WPB;
        if (t < FULL_TILES) {
            if (tn < FULL_TILES) {
                issue_tile(tn, cur ^ 1u);
                asm volatile("s_wait_asynccnt 0x3" ::: "memory"); // tile t landed
            } else {
                asm volatile("s_wait_asynccnt 0x0" ::: "memory");
            }
            float4 vi = stage[wave][cur][0][lane];
            float4 vp = stage[wave][cur][1][lane];
            float4 vo = stage[wave][cur][2][lane];
            acc += pair_diff(vi, vp, vo);
            cur ^= 1u;
        } else {
            // tail tile: 16 remaining float4 groups, direct global loads
            if (lane < 16u) {
                const unsigned g = gbase + t * 32u + lane;
                acc += pair_diff(img[g], proj[g], org[g]);
            }
        }
        t = tn;
    }

    // ---- wave32 reduction via V_WMMA_F32_16X16X4_F32 -----------------------
    // A vgpr0 holds acc (K=0 / K=2 slots), vgpr1 zero; B = all-ones.
    // D[m][n] = acc[m] + acc[m+16]; each lane sums its 8 D regs (8 rows of one
    // column), shfl_xor(16) pairs the two half-column owners -> full 32-lane sum.
    v2f a; a[0] = acc;  a[1] = 0.0f;
    v2f b; b[0] = 1.0f; b[1] = 1.0f;
    v8f c = {};
    v8f d = __builtin_amdgcn_wmma_f32_16x16x4_f32(
        /*neg_a=*/false, a, /*neg_b=*/false, b,
        /*c_mod=*/(short)0, c, /*reuse_a=*/false, /*reuse_b=*/false);
    float r = ((d[0] + d[1]) + (d[2] + d[3])) + ((d[4] + d[5]) + (d[6] + d[7]));
    r += __shfl_xor(r, 16, 32);

    if (lane == 0) wsum[wave] = r;
    __syncthreads();
    if (tid == 0) {
        float s = 0.0f;
#pragma unroll
        for (int w = 0; w < WAVES; ++w) s += wsum[w];
        partials[blockIdx.x] = s;   // every block writes exactly one slot
    }
}

__global__ __launch_bounds__(64)
void matchloss_final_kernel(const float* __restrict__ partials,
                            float* __restrict__ out) {
    __shared__ float fb[NBATCH];
    const int b = threadIdx.x;
    float s = 0.0f;
#pragma unroll 4
    for (int i = 0; i < BLK_PER_B; ++i) s += partials[b * BLK_PER_B + i];

    const float diff = s * (1.0f / ((float)NPTS * DENOM));
    const float rneg = fmaxf(-diff, 0.0f);
    const float rpos = fmaxf(diff, 0.0f);
    const float f    = ((1.0f + 5.0f * rneg) * diff + 5.0f * rpos)
                     / (1.0f + fabsf(diff));
    fb[b] = f;
    __syncthreads();
    if (b == 0) {
        float tot = 0.0f;
#pragma unroll
        for (int i = 0; i < NBATCH; ++i) tot += fb[i];
        out[0] = 1000.0f * tot * (1.0f / (float)NBATCH);
    }
}

extern "C" void kernel_launch(void* const* d_in, const int* in_sizes, int n_in,
                              void* d_out, int out_size, void* d_ws, size_t ws_size,
                              hipStream_t stream) {
    (void)in_sizes; (void)n_in; (void)out_size; (void)ws_size;
    const float4* img  = (const float4*)d_in[0];  // img_points        [64,100000,2] f32
    const float4* proj = (const float4*)d_in[1];  // projected_lid_... [64,100000,2] f32
    // d_in[2] = mat_reg_loss : unused by the reference
    const float4* org  = (const float4*)d_in[3];  // lid_cents_proj_Org
    float* partials = (float*)d_ws;               // 2048 floats, all rewritten each call
    float* out      = (float*)d_out;

    matchloss_partial_kernel<<<NBATCH * BLK_PER_B, THREADS, 0, stream>>>(
        img, proj, org, partials);
    matchloss_final_kernel<<<1, 64, 0, stream>>>(partials, out);
}